// GraphConv_13649406066773
// MI455X (gfx1250) — compile-verified
//
#include <hip/hip_runtime.h>
#include <hip/hip_bf16.h>

typedef float v8f __attribute__((ext_vector_type(8)));
typedef float v2f __attribute__((ext_vector_type(2)));

#define IN_DIM  128
#define OUT_DIM 128

// ---------------------------------------------------------------------------
// Kernel 1: zero the aggregation workspace (atomics accumulate into it)
// ---------------------------------------------------------------------------
__global__ void gc_zero_kernel(float4* __restrict__ p, long n4) {
    long i = (long)blockIdx.x * blockDim.x + threadIdx.x;
    if (i < n4) {
        p[i] = make_float4(0.f, 0.f, 0.f, 0.f);
    }
}

// ---------------------------------------------------------------------------
// Kernel 2: edge scatter.  One wave (32 lanes) per edge; each lane handles a
// contiguous float4 of the 128-wide feature row.  feat (25.6MB) is L2
// resident so random gathers are cheap; adds go to L2 atomic units.
// ---------------------------------------------------------------------------
__global__ void gc_scatter_kernel(const float* __restrict__ feat,
                                  const float* __restrict__ affine,
                                  const int*   __restrict__ src,
                                  const int*   __restrict__ dst,
                                  float*       __restrict__ agg,
                                  int n_edges) {
    long gid  = (long)blockIdx.x * blockDim.x + threadIdx.x;
    int  e    = (int)(gid >> 5);
    int  lane = (int)(gid & 31);
    if (e >= n_edges) return;

    int   s = src[e];
    int   d = dst[e];
    float a = affine[e];

    const float4 f = *(const float4*)(feat + (size_t)s * IN_DIM + lane * 4);
    float* o = agg + (size_t)d * IN_DIM + lane * 4;
    atomicAdd(o + 0, f.x * a);
    atomicAdd(o + 1, f.y * a);
    atomicAdd(o + 2, f.z * a);
    atomicAdd(o + 3, f.w * a);
}

// ---------------------------------------------------------------------------
// Kernel 3: out = relu(concat(feat, agg) @ W + b) with V_WMMA_F32_16X16X4_F32.
// Block = 256 threads = 8 waves; block owns 16 rows (M-tile), each wave owns a
// 16-wide N-tile (8 * 16 = 128 = OUT_DIM).  A tile (16x256) staged in LDS with
// +4 padding so the per-wave ds_load_2addr_b32 A-fragment read spans 64 banks.
// ---------------------------------------------------------------------------
__global__ void __launch_bounds__(256, 4)
gc_gemm_kernel(const float* __restrict__ feat,
               const float* __restrict__ agg,
               const float* __restrict__ W,     // [256, 128] row-major
               const float* __restrict__ bias,  // [128]
               float*       __restrict__ out,   // [n_nodes, 128]
               int n_nodes) {
    __shared__ float lds_a[16][256 + 4];

    const int tid   = threadIdx.x;
    const int mBase = blockIdx.x * 16;

    // Cooperative load of A tile: rows mBase..mBase+15, cols 0..255
    // col < 128 -> feat, col >= 128 -> agg
    for (int i = tid; i < 16 * 256; i += 256) {
        int row = i >> 8;
        int col = i & 255;
        int m   = mBase + row;
        float v = 0.f;
        if (m < n_nodes) {
            v = (col < IN_DIM) ? feat[(size_t)m * IN_DIM + col]
                               : agg [(size_t)m * IN_DIM + (col - IN_DIM)];
        }
        lds_a[row][col] = v;
    }
    __syncthreads();

    const int wave  = tid >> 5;
    const int lane  = tid & 31;
    const int nl    = lane & 15;   // A: M row within tile / B,C: N col within tile
    const int hi    = lane >> 4;   // 0: K=k0,k0+1   1: K=k0+2,k0+3
    const int nBase = wave * 16;

    v8f c = {};

    // A fragment pointer: lds_a[M = nl][k0 + hi*2 .. +1]
    const float* ap = &lds_a[nl][hi * 2];
    // B fragment pointer: W[(k0 + hi*2) * 128 + nBase + nl], and +128 for K+1
    const float* wp = W + (size_t)(hi * 2) * OUT_DIM + nBase + nl;

#pragma unroll 8
    for (int k0 = 0; k0 < 256; k0 += 4) {
        v2f a;
        a.x = ap[0];
        a.y = ap[1];
        v2f bm;
        bm.x = wp[0];
        bm.y = wp[OUT_DIM];
        c = __builtin_amdgcn_wmma_f32_16x16x4_f32(
                /*neg_a=*/false, a, /*neg_b=*/false, bm,
                /*c_mod=*/(short)0, c, /*reuse_a=*/false, /*reuse_b=*/false);
        ap += 4;
        wp += 4 * OUT_DIM;
    }

    // Epilogue: C layout — VGPR v: lanes 0-15 -> M=v, lanes 16-31 -> M=v+8; N=nl
    // Bounds are uniform per block (max row touched = mBase+15), so use one
    // scalar branch -> fast path is 8 straight global_store_b32.
    float bv = bias[nBase + nl];
    if (mBase + 16 <= n_nodes) {
        float* op = out + (size_t)(mBase + hi * 8) * OUT_DIM + nBase + nl;
#pragma unroll
        for (int v = 0; v < 8; ++v) {
            float val = c[v] + bv;
            op[(size_t)v * OUT_DIM] = val > 0.f ? val : 0.f;
        }
    } else {
#pragma unroll
        for (int v = 0; v < 8; ++v) {
            int m = mBase + v + hi * 8;
            if (m < n_nodes) {
                float val = c[v] + bv;
                out[(size_t)m * OUT_DIM + nBase + nl] = val > 0.f ? val : 0.f;
            }
        }
    }
}

// ---------------------------------------------------------------------------
extern "C" void kernel_launch(void* const* d_in, const int* in_sizes, int n_in,
                              void* d_out, int out_size, void* d_ws, size_t ws_size,
                              hipStream_t stream) {
    const float* feat   = (const float*)d_in[0];
    const float* affine = (const float*)d_in[1];
    const float* W      = (const float*)d_in[2];
    const float* bias   = (const float*)d_in[3];
    const int*   src    = (const int*)d_in[4];
    const int*   dst    = (const int*)d_in[5];

    const int n_nodes = in_sizes[0] / IN_DIM;
    const int n_edges = in_sizes[4];

    float* agg = (float*)d_ws;               // [n_nodes, 128] scratch
    float* out = (float*)d_out;              // [n_nodes, 128]

    // 1) zero agg
    {
        long n4 = (long)n_nodes * IN_DIM / 4;
        int blocks = (int)((n4 + 255) / 256);
        gc_zero_kernel<<<blocks, 256, 0, stream>>>((float4*)agg, n4);
    }

    // 2) edge scatter (one wave per edge)
    {
        long total  = (long)n_edges * 32;
        int  blocks = (int)((total + 255) / 256);
        gc_scatter_kernel<<<blocks, 256, 0, stream>>>(feat, affine, src, dst,
                                                      agg, n_edges);
    }

    // 3) WMMA GEMM + bias + relu
    {
        int blocks = (n_nodes + 15) / 16;
        gc_gemm_kernel<<<blocks, 256, 0, stream>>>(feat, agg, W, bias, out,
                                                   n_nodes);
    }
}